// LowRankLayer_44306882626256
// MI455X (gfx1250) — compile-verified
//
#include <hip/hip_runtime.h>

// LowRankLayer: Y = (U^T V) X  ==  U^T (V X),  L=8192, B=4096, RANK=8, all f32.
// Memory-bound: stream X once, stream Y once (~256 MiB @ 23.3 TB/s ~ 11.5us).
// Rank reduction done on the matrix pipe with V_WMMA_F32_16X16X4_F32.

typedef float v2f __attribute__((ext_vector_type(2)));
typedef float v8f __attribute__((ext_vector_type(8)));

#define L_DIM   8192
#define B_DIM   4096
#define RANK    8
#define LCHUNK  1024
#define NCHUNK  (L_DIM / LCHUNK)          // 8
#define LDS_ROWS 16                       // rows 8..15 zero-padded (A rows 8..15)
#define LDS_STRIDE (LCHUNK + 2)           // pad: row stride == 2 mod 64 banks

#define P1_COLS_PER_WAVE  16
#define P1_WAVES          8               // 256 threads = 8 wave32
#define P1_COLS_PER_BLOCK (P1_COLS_PER_WAVE * P1_WAVES)  // 128

#define P2_TILES_PER_WAVE 4               // 4 x 16 = 64 cols per wave

// ---------------------------------------------------------------------------
// Pass 1: Tpart[chunk][r][c] = sum_{l in chunk} V[r][l] * X[l][c]
// One wave = 16 columns; WMMA A = Vpad(16x4, rows 8..15 = 0), B = X(4x16).
// ---------------------------------------------------------------------------
__global__ __launch_bounds__(256) void lr_pass1(const float* __restrict__ X,
                                                const float* __restrict__ V,
                                                float* __restrict__ Tpart) {
  __shared__ float Vs[LDS_ROWS * LDS_STRIDE];
  const int tid  = threadIdx.x;
  const int wave = tid >> 5;
  const int lane = tid & 31;
  const int m    = lane & 15;     // matrix row (A) / column index (B,D)
  const int h    = lane >> 4;     // lane-half selects K in {2h, 2h+1}
  const int chunk = blockIdx.y;
  const int lbase = chunk * LCHUNK;
  const int c0    = blockIdx.x * P1_COLS_PER_BLOCK + wave * P1_COLS_PER_WAVE;

  // Stage Vpad[0:16][lbase : lbase+LCHUNK] into LDS; rows 8..15 are zeros so
  // lanes m>=8 feed true zeros to the WMMA (no per-iteration selects).
  for (int i = tid; i < LDS_ROWS * LCHUNK; i += 256) {
    const int r = i >> 10;              // i / LCHUNK
    const int l = i & (LCHUNK - 1);
    Vs[r * LDS_STRIDE + l] = (r < RANK) ? V[(size_t)r * L_DIM + lbase + l] : 0.0f;
  }
  __syncthreads();

  v8f acc = {};
  // Per-lane streaming pointers.
  const float* xp  = X + (size_t)(lbase + 2 * h) * B_DIM + (c0 + m);
  const float* vsp = &Vs[m * LDS_STRIDE + 2 * h];

#pragma unroll 4
  for (int kk = 0; kk < LCHUNK / 4; ++kk) {
    // A[m][K]: vgpr0 -> K=2h, vgpr1 -> K=2h+1 (contiguous l => one b64 LDS load)
    const float2 av = *(const float2*)vsp;
    v2f a;
    a.x = av.x;
    a.y = av.y;
    // B[K][n]: rows lbase+4kk+2h and +1, columns c0..c0+15 (NT: no reuse of X)
    v2f b;
    b.x = __builtin_nontemporal_load(xp);
    b.y = __builtin_nontemporal_load(xp + B_DIM);
    acc = __builtin_amdgcn_wmma_f32_16x16x4_f32(false, a, false, b,
                                                (short)0, acc, false, false);
    vsp += 4;
    xp  += (size_t)4 * B_DIM;
  }

  // D rows 0..7 live in vgprs 0..7 of lanes 0..15 (rows 8..15 are zeros).
  if (h == 0) {
    float* tp = Tpart + ((size_t)chunk * RANK) * B_DIM + (c0 + m);
#pragma unroll
    for (int j = 0; j < RANK; ++j)
      tp[(size_t)j * B_DIM] = acc[j];
  }
}

// ---------------------------------------------------------------------------
// Reduce: T[r][c] = sum_chunk Tpart[chunk][r][c]   (fixed order -> deterministic)
// ---------------------------------------------------------------------------
__global__ __launch_bounds__(256) void lr_reduce(const float* __restrict__ Tpart,
                                                 float* __restrict__ T) {
  const int i = blockIdx.x * 256 + threadIdx.x;   // over RANK*B_DIM = 32768
  if (i < RANK * B_DIM) {
    float s = 0.0f;
#pragma unroll
    for (int c = 0; c < NCHUNK; ++c)
      s += Tpart[(size_t)c * RANK * B_DIM + i];
    T[i] = s;
  }
}

// ---------------------------------------------------------------------------
// Pass 2: Y[l][c] = sum_r U[r][l] * T[r][c]
// One wave = 16 rows x 64 columns; A tiles (U^T) loop-invariant in registers.
// ---------------------------------------------------------------------------
__global__ __launch_bounds__(256) void lr_pass2(const float* __restrict__ U,
                                                const float* __restrict__ T,
                                                float* __restrict__ Y) {
  const int tid  = threadIdx.x;
  const int wave = tid >> 5;
  const int lane = tid & 31;
  const int m    = lane & 15;
  const int h    = lane >> 4;
  const int l0    = blockIdx.y * 16;
  const int cbase = (blockIdx.x * P1_WAVES + wave) * (P2_TILES_PER_WAVE * 16);

  // A[m][K] = U[rank][l0+m]; chunk0 ranks {2h, 2h+1}, chunk1 ranks {4+2h, 5+2h}
  v2f a0, a1;
  a0.x = U[(size_t)(0 + 2 * h) * L_DIM + l0 + m];
  a0.y = U[(size_t)(1 + 2 * h) * L_DIM + l0 + m];
  a1.x = U[(size_t)(4 + 2 * h) * L_DIM + l0 + m];
  a1.y = U[(size_t)(5 + 2 * h) * L_DIM + l0 + m];

#pragma unroll
  for (int t = 0; t < P2_TILES_PER_WAVE; ++t) {
    const int c0 = cbase + t * 16;
    // B[K][n] = T[rank][c0+n]  (T is 128 KiB -> L2 resident)
    v2f b0, b1;
    b0.x = T[(size_t)(0 + 2 * h) * B_DIM + c0 + m];
    b0.y = T[(size_t)(1 + 2 * h) * B_DIM + c0 + m];
    b1.x = T[(size_t)(4 + 2 * h) * B_DIM + c0 + m];
    b1.y = T[(size_t)(5 + 2 * h) * B_DIM + c0 + m];

    v8f acc = {};
    acc = __builtin_amdgcn_wmma_f32_16x16x4_f32(false, a0, false, b0,
                                                (short)0, acc, false, false);
    acc = __builtin_amdgcn_wmma_f32_16x16x4_f32(false, a1, false, b1,
                                                (short)0, acc, false, false);

    // D[j] holds output row l0 + j + 8h, column c0 + m. NT streaming stores.
    float* yp = Y + (size_t)(l0 + 8 * h) * B_DIM + c0 + m;
#pragma unroll
    for (int j = 0; j < 8; ++j)
      __builtin_nontemporal_store(acc[j], yp + (size_t)j * B_DIM);
  }
}

// ---------------------------------------------------------------------------
extern "C" void kernel_launch(void* const* d_in, const int* in_sizes, int n_in,
                              void* d_out, int out_size, void* d_ws, size_t ws_size,
                              hipStream_t stream) {
  const float* X = (const float*)d_in[0];   // inputs [L, B]
  const float* U = (const float*)d_in[1];   // [RANK, L]
  const float* V = (const float*)d_in[2];   // [RANK, L]
  float* Y = (float*)d_out;                 // [L, B]

  // Workspace: Tpart (NCHUNK*RANK*B = 1 MiB) followed by T (RANK*B = 128 KiB).
  float* Tpart = (float*)d_ws;
  float* T     = Tpart + (size_t)NCHUNK * RANK * B_DIM;

  dim3 g1(B_DIM / P1_COLS_PER_BLOCK, NCHUNK);             // 32 x 8
  lr_pass1<<<g1, 256, 0, stream>>>(X, V, Tpart);

  lr_reduce<<<(RANK * B_DIM) / 256, 256, 0, stream>>>(Tpart, T);

  dim3 g2(B_DIM / (P1_WAVES * P2_TILES_PER_WAVE * 16),    // 8
          L_DIM / 16);                                    // 512
  lr_pass2<<<g2, 256, 0, stream>>>(U, T, Y);
}